// InceptionGCNBlock_16338055594707
// MI455X (gfx1250) — compile-verified
//
#include <hip/hip_runtime.h>

typedef __attribute__((ext_vector_type(2))) float v2f;
typedef __attribute__((ext_vector_type(8))) float v8f;

#define NN 50000
#define NE 800000
#define DD 128

// -----------------------------------------------------------------------------
// GEMM: T1 = h @ W ; T2 = h @ SW + bias      h:[NN,128], W/SW:[128,128] row-major
// One block = 16 rows of h; 8 waves, wave w owns output columns [16w,16w+16).
// fp32 WMMA 16x16x4, A tile staged in LDS (padded stride 132 -> conflict-free
// ds_load_b64 of (a[m][k], a[m][k+1]) per K-step).
// -----------------------------------------------------------------------------
__global__ __launch_bounds__(256) void gcn_gemm(
    const float* __restrict__ h, const float* __restrict__ W,
    const float* __restrict__ SW, const float* __restrict__ bias,
    float* __restrict__ T1, float* __restrict__ T2)
{
    __shared__ float ah[16][132];   // 16 rows x 128 cols, padded to 132
    const int m0 = blockIdx.x * 16;

    // cooperative float4 load of the 16x128 A tile
    for (int t = threadIdx.x; t < 512; t += 256) {
        const int r  = t >> 5;   // 0..15
        const int c4 = t & 31;   // 0..31
        const float4 v = *(const float4*)(h + (size_t)(m0 + r) * DD + c4 * 4);
        *(float4*)&ah[r][c4 * 4] = v;
    }
    __syncthreads();

    const int wave = threadIdx.x >> 5;
    const int lane = threadIdx.x & 31;
    const int nl   = lane & 15;
    const int hi   = (lane >= 16) ? 1 : 0;
    const int n0   = wave * 16;
    const int colN = n0 + nl;

    v8f accW = {};
    v8f accS = {};

    for (int k = 0; k < DD; k += 4) {
        const int kk = k + 2 * hi;
        // A fragment (ISA 16x4 f32 layout): lanes 0-15 -> K=k,k+1 ; 16-31 -> K=k+2,k+3
        v2f a;
        a.x = ah[nl][kk];
        a.y = ah[nl][kk + 1];
        // B fragment (4x16, transposed-analogue layout)
        v2f bw, bs;
        bw.x = W [(size_t)kk       * DD + colN];
        bw.y = W [(size_t)(kk + 1) * DD + colN];
        bs.x = SW[(size_t)kk       * DD + colN];
        bs.y = SW[(size_t)(kk + 1) * DD + colN];

        accW = __builtin_amdgcn_wmma_f32_16x16x4_f32(
            false, a, false, bw, (short)0, accW, false, false);
        accS = __builtin_amdgcn_wmma_f32_16x16x4_f32(
            false, a, false, bs, (short)0, accS, false, false);
    }

    // C/D layout: VGPR r -> (M = r + 8*hi, N = colN)
    const float bb = bias[colN];
#pragma unroll
    for (int r = 0; r < 8; ++r) {
        const int m = m0 + r + 8 * hi;
        T1[(size_t)m * DD + colN] = accW[r];
        T2[(size_t)m * DD + colN] = accS[r] + bb;   // agg init = h@SW + b
    }
}

// -----------------------------------------------------------------------------
// SpMM scatter: T2[row[e]] += ew[e] * T1[col[e]]   (one wave per edge,
// lane handles a float4 slice of the 128-wide row; f32 global atomics)
// -----------------------------------------------------------------------------
__global__ __launch_bounds__(256) void gcn_spmm(
    const float* __restrict__ T1, float* __restrict__ T2,
    const int* __restrict__ row, const int* __restrict__ col,
    const float* __restrict__ ew)
{
    const int gwave  = (int)((blockIdx.x * 256u + threadIdx.x) >> 5);
    const int lane   = threadIdx.x & 31;
    const int nwaves = (int)((gridDim.x * 256u) >> 5);

    for (int e = gwave; e < NE; e += nwaves) {
        const int   r = row[e];
        const int   c = col[e];
        const float w = ew[e];
        const float4 v = *(const float4*)(T1 + (size_t)c * DD + lane * 4);
        float* dst = T2 + (size_t)r * DD + lane * 4;
        atomicAdd(dst + 0, v.x * w);
        atomicAdd(dst + 1, v.y * w);
        atomicAdd(dst + 2, v.z * w);
        atomicAdd(dst + 3, v.w * w);
    }
}

// -----------------------------------------------------------------------------
// Epilogue: xout = xprev + relu(BN_eval(agg))
// -----------------------------------------------------------------------------
__global__ __launch_bounds__(256) void gcn_bn_relu_add(
    const float* __restrict__ xprev, const float* __restrict__ agg,
    float* __restrict__ xout,
    const float* __restrict__ gamma, const float* __restrict__ beta,
    const float* __restrict__ rmean, const float* __restrict__ rvar)
{
    const int i = (int)(blockIdx.x * 256u + threadIdx.x);
    if (i < NN * DD) {
        const int c = i & (DD - 1);
        float v = (agg[i] - rmean[c]) * rsqrtf(rvar[c] + 1e-5f) * gamma[c] + beta[c];
        v = fmaxf(v, 0.0f);
        xout[i] = xprev[i] + v;
    }
}

// -----------------------------------------------------------------------------
// Host launcher.  Only layers 0, 2, 5 affect the output (inner layers that are
// overwritten before use are dead code).  ws layout: T1 | T2 | xbuf, 25.6MB each.
// -----------------------------------------------------------------------------
extern "C" void kernel_launch(void* const* d_in, const int* in_sizes, int n_in,
                              void* d_out, int out_size, void* d_ws, size_t ws_size,
                              hipStream_t stream) {
    (void)in_sizes; (void)n_in; (void)out_size; (void)ws_size;

    const float* x0    = (const float*)d_in[0];
    const int*   row   = (const int*)  d_in[1];
    const int*   col   = (const int*)  d_in[2];
    const float* ew    = (const float*)d_in[3];
    const float* W     = (const float*)d_in[4];
    const float* SW    = (const float*)d_in[5];
    const float* bias  = (const float*)d_in[6];
    const float* gamma = (const float*)d_in[7];
    const float* beta  = (const float*)d_in[8];
    const float* rmean = (const float*)d_in[9];
    const float* rvar  = (const float*)d_in[10];
    float* out = (float*)d_out;

    const size_t mat = (size_t)NN * DD;
    float* T1   = (float*)d_ws;
    float* T2   = T1 + mat;
    float* xbuf = T2 + mat;

    const int    lyr[3]  = {0, 2, 5};
    const float* xin[3]  = {x0,   xbuf, out};
    float*       xout[3] = {xbuf, out,  out};

    for (int s = 0; s < 3; ++s) {
        const int l = lyr[s];
        gcn_gemm<<<NN / 16, 256, 0, stream>>>(
            xin[s], W + (size_t)l * DD * DD, SW + (size_t)l * DD * DD,
            bias + (size_t)l * DD, T1, T2);
        gcn_spmm<<<4096, 256, 0, stream>>>(T1, T2, row, col, ew);
        gcn_bn_relu_add<<<(NN * DD + 255) / 256, 256, 0, stream>>>(
            xin[s], T2, xout[s],
            gamma + (size_t)l * DD, beta + (size_t)l * DD,
            rmean + (size_t)l * DD, rvar + (size_t)l * DD);
    }
}